// Fastformer_Graph_Lite_90323162235286
// MI455X (gfx1250) — compile-verified
//
#include <hip/hip_runtime.h>
#include <hip/hip_bf16.h>

typedef float v2f __attribute__((ext_vector_type(2)));
typedef float v8f __attribute__((ext_vector_type(8)));

#define D 128
#define HIST 50
#define CAND 20

// -------------------------------------------------------------------------
// Kernel 1: news_emb[N,128] = x[N,300] @ W_enc[300,128]
// One wave (32 threads) per 16x16 output tile, K stepped by 4 via
// V_WMMA_F32_16X16X4_F32 (fp32 matrix path; reference is fp32).
// -------------------------------------------------------------------------
__global__ __launch_bounds__(32) void encoder_gemm(
    const float* __restrict__ x, const float* __restrict__ W,
    float* __restrict__ out, int K /*=300*/) {
  const int lane = threadIdx.x;            // 0..31
  const int tile = blockIdx.x;             // (N/16)*8 tiles
  const int mt = tile >> 3;                // M tile index
  const int nt = tile & 7;                 // N tile index (128/16 = 8)
  const int arow = mt * 16 + (lane & 15);  // A row for this lane
  const int col  = nt * 16 + (lane & 15);  // B/D column for this lane
  const int kh   = (lane >> 4) * 2;        // lanes 16-31 hold K+2,K+3

  const float* __restrict__ xr = x + (size_t)arow * K;
  v8f acc = {};
  for (int k = 0; k < K; k += 4) {
    v2f a, b;
    a.x = xr[k + kh + 0];
    a.y = xr[k + kh + 1];
    b.x = W[(size_t)(k + kh + 0) * D + col];
    b.y = W[(size_t)(k + kh + 1) * D + col];
    acc = __builtin_amdgcn_wmma_f32_16x16x4_f32(
        false, a, false, b, (short)0, acc, false, false);
  }
  // D layout: VGPR r -> M = r (lanes 0-15) or r+8 (lanes 16-31), N = col
  const int mbase = mt * 16 + (lane >> 4) * 8;
#pragma unroll
  for (int r = 0; r < 8; ++r)
    out[(size_t)(mbase + r) * D + col] = acc[r];
}

// -------------------------------------------------------------------------
// Kernel 2: edge scatter-add.  idx -> (edge, dim); 128 consecutive threads
// share one edge -> coalesced gather of news_emb[src] and coalesced
// global_atomic_add_f32 into agg[dst] (L2-resident: agg is 51 MB < 192 MB).
// -------------------------------------------------------------------------
__global__ void edge_agg(const int* __restrict__ ei,
                         const float* __restrict__ news,
                         float* __restrict__ agg, float* __restrict__ deg,
                         long long E) {
  long long idx = (long long)blockIdx.x * blockDim.x + threadIdx.x;
  long long e = idx >> 7;
  int d = (int)(idx & (D - 1));
  if (e >= E) return;
  int s = ei[e];       // edge_index[0][e] (src)
  int t = ei[E + e];   // edge_index[1][e] (dst)
  atomicAdd(&agg[(size_t)t * D + d], news[(size_t)s * D + d]);
  if (d == 0) atomicAdd(&deg[t], 1.0f);
}

// -------------------------------------------------------------------------
// Kernel 3: xx = news_emb + (agg / max(deg,1)) @ W_gnn   (K = 128)
// Same fp32 WMMA tiling; degree normalization folded into the A fragment.
// -------------------------------------------------------------------------
__global__ __launch_bounds__(32) void gnn_gemm_add(
    const float* __restrict__ agg, const float* __restrict__ deg,
    const float* __restrict__ Wg, const float* __restrict__ news,
    float* __restrict__ xx) {
  const int lane = threadIdx.x;
  const int tile = blockIdx.x;
  const int mt = tile >> 3;
  const int nt = tile & 7;
  const int arow = mt * 16 + (lane & 15);
  const int col  = nt * 16 + (lane & 15);
  const int kh   = (lane >> 4) * 2;

  const float dinv = 1.0f / fmaxf(deg[arow], 1.0f);
  const float* __restrict__ ar = agg + (size_t)arow * D;
  v8f acc = {};
  for (int k = 0; k < D; k += 4) {
    v2f a, b;
    a.x = ar[k + kh + 0] * dinv;
    a.y = ar[k + kh + 1] * dinv;
    b.x = Wg[(size_t)(k + kh + 0) * D + col];
    b.y = Wg[(size_t)(k + kh + 1) * D + col];
    acc = __builtin_amdgcn_wmma_f32_16x16x4_f32(
        false, a, false, b, (short)0, acc, false, false);
  }
  const int mbase = mt * 16 + (lane >> 4) * 8;
#pragma unroll
  for (int r = 0; r < 8; ++r) {
    size_t o = (size_t)(mbase + r) * D + col;
    xx[o] = news[o] + acc[r];
  }
}

// -------------------------------------------------------------------------
// Kernel 4: fused user-vector (masked mean over history) + candidate scores.
// One 128-thread block per user; user_vec staged in LDS; each of the 4
// waves reduces dot products for candidates c = wave, wave+4, ...
// -------------------------------------------------------------------------
__global__ __launch_bounds__(128) void user_score(
    const float* __restrict__ xx, const int* __restrict__ hist,
    const int* __restrict__ cand, float* __restrict__ scores) {
  const int b = blockIdx.x;
  const int t = threadIdx.x;      // 0..127 == dim
  __shared__ float uv[D];

  float sum = 0.0f, cnt = 0.0f;
  for (int h = 0; h < HIST; ++h) {
    int nid = hist[b * HIST + h];
    if (nid != 0) {
      sum += xx[(size_t)nid * D + t];
      cnt += 1.0f;
    }
  }
  uv[t] = sum / fmaxf(cnt, 1e-9f);
  __syncthreads();

  const int wave = t >> 5, lane = t & 31;
  for (int c = wave; c < CAND; c += 4) {
    int nid = cand[b * CAND + c];
    const float* __restrict__ ce = xx + (size_t)nid * D;
    float p = 0.0f;
#pragma unroll
    for (int d = lane; d < D; d += 32) p += ce[d] * uv[d];
#pragma unroll
    for (int off = 16; off; off >>= 1) p += __shfl_xor(p, off, 32);
    if (lane == 0) scores[b * CAND + c] = p;
  }
}

// -------------------------------------------------------------------------
extern "C" void kernel_launch(void* const* d_in, const int* in_sizes, int n_in,
                              void* d_out, int out_size, void* d_ws, size_t ws_size,
                              hipStream_t stream) {
  const float* x      = (const float*)d_in[0];   // [N,300]
  // d_in[1] = n_id == arange(N): identity scatter, unused
  const int*   ei     = (const int*)  d_in[2];   // [2,E]
  const int*   hist   = (const int*)  d_in[3];   // [B,50]
  const int*   cand   = (const int*)  d_in[4];   // [B,20]
  const float* W_enc  = (const float*)d_in[5];   // [300,128]
  const float* W_gnn  = (const float*)d_in[6];   // [128,128]
  float*       scores = (float*)d_out;           // [B,20]

  const int       N = in_sizes[1];               // 100000 (divisible by 16)
  const int       K = in_sizes[0] / N;           // 300
  const long long E = (long long)in_sizes[2] / 2;
  const int       B = in_sizes[3] / HIST;        // 4096

  float* news = (float*)d_ws;                    // [N,128]
  float* agg  = news + (size_t)N * D;            // [N,128]
  float* deg  = agg  + (size_t)N * D;            // [N]
  float* xx   = deg  + (size_t)N;                // [N,128]

  // Accumulators must be zero every call (graph-capture-safe memset nodes).
  hipMemsetAsync(agg, 0, (size_t)N * D * sizeof(float), stream);
  hipMemsetAsync(deg, 0, (size_t)N * sizeof(float), stream);

  const int tiles = (N / 16) * (D / 16);         // 6250 * 8
  encoder_gemm<<<tiles, 32, 0, stream>>>(x, W_enc, news, K);

  long long work = E * D;
  int eblocks = (int)((work + 255) / 256);
  edge_agg<<<eblocks, 256, 0, stream>>>(ei, news, agg, deg, E);

  gnn_gemm_add<<<tiles, 32, 0, stream>>>(agg, deg, W_gnn, news, xx);

  user_score<<<B, 128, 0, stream>>>(xx, hist, cand, scores);
}